// FS4Ddeq_model_24292335026648
// MI455X (gfx1250) — compile-verified
//
#include <hip/hip_runtime.h>
#include <hip/hip_bf16.h>
#include <math.h>

// Problem constants (match reference)
#define BB 4
#define LL 2048
#define CC 256
#define NN2 32
#define NITER 12
#define SS ((size_t)BB * LL * CC)   // 2,097,152 elements

typedef float v2f __attribute__((ext_vector_type(2)));
typedef float v8f __attribute__((ext_vector_type(8)));

// ---------------------------------------------------------------------------
// S4D discretization: per (c,n) compute w = exp(dt*A) and 2*Cd (ZOH)
// P[i] = (w_re, w_im, 2*Cd_re, -2*Cd_im)  so  y_contrib = P.z*s_re + P.w*s_im
// ---------------------------------------------------------------------------
__global__ void s4d_params_kernel(const float* __restrict__ log_dt,
                                  const float* __restrict__ Are,
                                  const float* __restrict__ Aim,
                                  const float* __restrict__ Cre,
                                  const float* __restrict__ Cim,
                                  float4* __restrict__ P) {
  int i = blockIdx.x * blockDim.x + threadIdx.x;
  if (i >= CC * NN2) return;
  int c = i / NN2;
  float dt  = expf(log_dt[c]);
  float ar  = -expf(Are[i]);      // Re(A)
  float ai  = Aim[i];             // Im(A)
  float dre = dt * ar, dim = dt * ai;
  float ew  = expf(dre);
  float cs  = cosf(dim), sn = sinf(dim);
  float wre = ew * cs, wim = ew * sn;
  // complex expm1(dtA) = expm1(dre)*cos + (cos-1) + i*exp(dre)*sin
  float e1r = expm1f(dre) * cs + (cs - 1.0f);
  float e1i = ew * sn;
  // q = expm1(dtA)/A = expm1 * conj(A)/|A|^2
  float den = ar * ar + ai * ai;
  float qre = (e1r * ar + e1i * ai) / den;
  float qim = (e1i * ar - e1r * ai) / den;
  float cr = Cre[i], ci = Cim[i];
  float cdre = cr * qre - ci * qim;
  float cdim = cr * qim + ci * qre;
  P[i] = make_float4(wre, wim, 2.0f * cdre, -2.0f * cdim);
}

// ---------------------------------------------------------------------------
// Row LayerNorm over C=256 (one block of 256 threads per row).
// Optional post-scale by sigmoid(*gate) (the DEQ feed gate).
// ---------------------------------------------------------------------------
__global__ void ln_rows_kernel(const float* __restrict__ in,
                               const float* __restrict__ g,
                               const float* __restrict__ bv,
                               float* __restrict__ out,
                               const float* __restrict__ gate) {
  int row = blockIdx.x, tid = threadIdx.x;
  size_t base = (size_t)row * CC;
  float v = in[base + tid];
  __shared__ float red[CC];
  red[tid] = v; __syncthreads();
  #pragma unroll
  for (int s = CC / 2; s > 0; s >>= 1) {
    if (tid < s) red[tid] += red[tid + s];
    __syncthreads();
  }
  float mu = red[0] * (1.0f / CC);
  __syncthreads();
  float d = v - mu;
  red[tid] = d * d; __syncthreads();
  #pragma unroll
  for (int s = CC / 2; s > 0; s >>= 1) {
    if (tid < s) red[tid] += red[tid + s];
    __syncthreads();
  }
  float var = red[0] * (1.0f / CC);
  float o = d * rsqrtf(var + 1e-5f) * g[tid] + bv[tid];
  if (gate) o *= 1.0f / (1.0f + expf(-gate[0]));
  out[base + tid] = o;
}

// ---------------------------------------------------------------------------
// Fused S4D depthwise causal conv as a diagonal-state recurrence.
// One wave32 per (b,c); lane = state index n.
// Per 32-step chunk: sequential state updates write per-lane contributions to
// a 32x33 LDS tile (conflict-free), then a transpose read gives each lane the
// 32 per-state contributions of *its* timestep to sum (replaces 5-10 shuffle
// reduces per step with 2 DS ops per step).
// Emits gelu(y + D*u) and (optionally) the G-kernel readout (feedback state).
// u = u0 - usub (usub nullable, for u1 = LN(x) - feed).
// ---------------------------------------------------------------------------
__global__ __launch_bounds__(32)
void s4d_conv_kernel(const float* __restrict__ u0,
                     const float* __restrict__ usub,
                     const float4* __restrict__ PK,
                     const float4* __restrict__ PG,
                     const float* __restrict__ Dp,
                     float* __restrict__ yout,
                     float* __restrict__ feedout) {
  __shared__ float tileK[32][33];
  __shared__ float tileG[32][33];
  int bc = blockIdx.x;
  int b = bc / CC, c = bc % CC;
  int lane = threadIdx.x;
  float4 pk = PK[c * NN2 + lane];
  float ga = 0.0f, gb = 0.0f;
  const bool hasG = (PG != nullptr);
  if (hasG) { float4 pg = PG[c * NN2 + lane]; ga = pg.z; gb = pg.w; }
  float D = Dp[0];
  float sre = 0.0f, sim = 0.0f;

  for (int t0 = 0; t0 < LL; t0 += 32) {
    size_t rowbase = (size_t)b * LL + t0;
    size_t myidx = (rowbase + lane) * CC + c;
    float uj = u0[myidx];
    if (usub) uj -= usub[myidx];
    if (t0 + 32 < LL) {
      // hide next chunk's stride-C gather (global_prefetch_b8)
      __builtin_prefetch(u0 + myidx + (size_t)32 * CC, 0, 1);
    }
    // ---- sequential recurrence over the 32 timesteps of this chunk ----
    #pragma unroll
    for (int j = 0; j < 32; ++j) {
      float ut = __shfl(uj, j, 32);                 // broadcast u[t0+j]
      float nre = pk.x * sre - pk.y * sim + ut;     // s = w*s + u
      float nim = pk.x * sim + pk.y * sre;
      sre = nre; sim = nim;
      tileK[j][lane] = pk.z * sre + pk.w * sim;     // 2*Re(Cd*s), this lane's n
      if (hasG) tileG[j][lane] = ga * sre + gb * sim;
    }
    __syncthreads();
    // ---- transpose-reduce: lane t sums contributions of timestep t ----
    float ysum = 0.0f, fsum = 0.0f;
    #pragma unroll
    for (int n = 0; n < 32; ++n) {
      ysum += tileK[lane][n];
      if (hasG) fsum += tileG[lane][n];
    }
    __syncthreads();
    float pre = ysum + D * uj;
    yout[myidx] = 0.5f * pre * (1.0f + erff(pre * 0.70710678118654752f)); // exact gelu
    if (feedout) feedout[myidx] = fsum;
  }
}

// ---------------------------------------------------------------------------
// fp32 GEMM with V_WMMA_F32_16X16X4_F32.
// One wave computes a 16(M) x 64(N) strip: the A fragment is loaded once per
// k-step and reused across 4 B fragments / 4 accumulators (4x less A traffic,
// 4 independent WMMA chains per k-step).  4 waves per 128-thread block.
// A frag (16x4): lanes 0-15 hold M=lane, v0=K0/v1=K1; lanes 16-31 K2/K3.
// B frag (4x16): lanes 0-15 hold N=lane, v0=K0/v1=K1; lanes 16-31 K2/K3.
// D/C (16x16):  vgpr r -> M = r + 8*half, N = lane&15.
// ---------------------------------------------------------------------------
__global__ __launch_bounds__(128)
void gemm_wmma_f32_kernel(const float* __restrict__ A,
                          const float* __restrict__ Bm,
                          const float* __restrict__ bias,
                          float* __restrict__ Cmat,
                          int M, int N, int K) {
  int ntn  = N >> 6;                          // 64-wide strips along N
  int tile = blockIdx.x * 4 + (threadIdx.x >> 5);
  int tm   = (tile / ntn) << 4;
  int tn0  = (tile % ntn) << 6;
  int lane = threadIdx.x & 31;
  int half = lane >> 4;
  int mr   = lane & 15;

  v8f acc0 = {}, acc1 = {}, acc2 = {}, acc3 = {};
  const float* Arow = A + (size_t)(tm + mr) * K;
  #pragma unroll 2
  for (int k0 = 0; k0 < K; k0 += 4) {
    int ka = k0 + 2 * half;
    v2f a;
    a[0] = Arow[ka];
    a[1] = Arow[ka + 1];
    const float* Bp = Bm + (size_t)ka * N + tn0 + mr;
    v2f b0, b1, b2, b3;
    b0[0] = Bp[0];  b0[1] = Bp[N];
    b1[0] = Bp[16]; b1[1] = Bp[N + 16];
    b2[0] = Bp[32]; b2[1] = Bp[N + 32];
    b3[0] = Bp[48]; b3[1] = Bp[N + 48];
    acc0 = __builtin_amdgcn_wmma_f32_16x16x4_f32(false, a, false, b0, (short)0, acc0, false, false);
    acc1 = __builtin_amdgcn_wmma_f32_16x16x4_f32(false, a, false, b1, (short)0, acc1, false, false);
    acc2 = __builtin_amdgcn_wmma_f32_16x16x4_f32(false, a, false, b2, (short)0, acc2, false, false);
    acc3 = __builtin_amdgcn_wmma_f32_16x16x4_f32(false, a, false, b3, (short)0, acc3, false, false);
  }
  #pragma unroll
  for (int r = 0; r < 8; ++r) {
    int m = tm + r + 8 * half;
    size_t rowoff = (size_t)m * N + tn0 + mr;
    Cmat[rowoff]      = acc0[r] + (bias ? bias[tn0 + mr]      : 0.0f);
    Cmat[rowoff + 16] = acc1[r] + (bias ? bias[tn0 + mr + 16] : 0.0f);
    Cmat[rowoff + 32] = acc2[r] + (bias ? bias[tn0 + mr + 32] : 0.0f);
    Cmat[rowoff + 48] = acc3[r] + (bias ? bias[tn0 + mr + 48] : 0.0f);
  }
}

// ---------------------------------------------------------------------------
// GLU combine: out = a * sigmoid(bgate) + res, with zg = [rows, 2C]
// ---------------------------------------------------------------------------
__global__ void glu_combine_kernel(const float* __restrict__ zg,
                                   const float* __restrict__ res,
                                   float* __restrict__ out) {
  size_t i = (size_t)blockIdx.x * blockDim.x + threadIdx.x;
  if (i >= SS) return;
  size_t r = i / CC, c = i % CC;
  float a  = zg[r * (2 * CC) + c];
  float bg = zg[r * (2 * CC) + CC + c];
  out[i] = a * (1.0f / (1.0f + expf(-bg))) + res[i];
}

// ---------------------------------------------------------------------------
// Per-(b,c) L2 norms over the sequence axis -> ratio; then mean.
// ---------------------------------------------------------------------------
__global__ void norm_ratio_kernel(const float* __restrict__ x,
                                  const float* __restrict__ z,
                                  float* __restrict__ ratios) {
  int bc = blockIdx.x;
  int b = bc / CC, c = bc % CC;
  int tid = threadIdx.x;
  float sx = 0.0f, sf = 0.0f;
  for (int l = tid; l < LL; l += 256) {
    size_t i = ((size_t)b * LL + l) * CC + c;
    float xv = x[i]; sx += xv * xv;
    float zv = z[i]; sf += zv * zv;
  }
  __shared__ float rx[256], rf[256];
  rx[tid] = sx; rf[tid] = sf; __syncthreads();
  #pragma unroll
  for (int s = 128; s > 0; s >>= 1) {
    if (tid < s) { rx[tid] += rx[tid + s]; rf[tid] += rf[tid + s]; }
    __syncthreads();
  }
  if (tid == 0) ratios[bc] = sqrtf(rf[0]) / (sqrtf(rx[0]) + 1e-8f);
}

__global__ void mean_ratio_kernel(const float* __restrict__ ratios,
                                  float* __restrict__ out) {
  int tid = threadIdx.x;
  float s = 0.0f;
  for (int i = tid; i < BB * CC; i += 256) s += ratios[i];
  __shared__ float red[256];
  red[tid] = s; __syncthreads();
  #pragma unroll
  for (int st = 128; st > 0; st >>= 1) {
    if (tid < st) red[tid] += red[tid + st];
    __syncthreads();
  }
  if (tid == 0) out[0] = red[0] / (float)(BB * CC);
}

// ---------------------------------------------------------------------------
// Orchestration
// ---------------------------------------------------------------------------
extern "C" void kernel_launch(void* const* d_in, const int* in_sizes, int n_in,
                              void* d_out, int out_size, void* d_ws, size_t ws_size,
                              hipStream_t stream) {
  (void)in_sizes; (void)n_in; (void)out_size; (void)ws_size;
  const float* x     = (const float*)d_in[0];
  const float* gamma = (const float*)d_in[1];
  const float* Wf    = (const float*)d_in[2];
  const float* bf    = (const float*)d_in[3];
  const float* lnf_g = (const float*)d_in[4];
  const float* lnf_b = (const float*)d_in[5];
  // p_in @6, p_mid @16, p_out @26 (p_out has 12 entries, G params at +10,+11)
  const int BIN = 6, BMID = 16, BOUT = 26;
  #define PIN(k)  ((const float*)d_in[BIN  + (k)])
  #define PMID(k) ((const float*)d_in[BMID + (k)])
  #define POUT(k) ((const float*)d_in[BOUT + (k)])

  // ---- workspace layout (floats) ----
  const size_t S = SS;
  const int    pn = CC * NN2;                 // 8192 float4 per kernel set
  float* w = (float*)d_ws;
  float4* P   = (float4*)w;                   // 4 sets * pn float4 = 131072 floats
  float* xln  = w + 4 * (size_t)pn * 4;
  float* zA   = xln  + S;
  float* zB   = zA   + S;
  float* tmp1 = zB   + S;
  float* feed = tmp1 + S;
  float* yact = feed + S;
  float* ynrm = yact + S;
  float* ubuf = ynrm + S;
  float* ybuf = ubuf + S;
  float* zg   = ybuf + S;                     // rows x 2C = 2S
  float* ratios = zg + 2 * S;

  const int ROWS = BB * LL;                   // 8192
  // 16x64 strips, 4 waves (strips) per 128-thread block
  const int G1 = (ROWS / 16) * (CC / 64) / 4;        // feed GEMM blocks (512)
  const int G2 = (ROWS / 16) * ((2 * CC) / 64) / 4;  // GLU GEMM blocks (1024)

  // ---- one-time (per call) precompute ----
  s4d_params_kernel<<<pn / 256, 256, 0, stream>>>(PIN(0), PIN(1), PIN(2), PIN(3), PIN(4), P + 0 * pn);
  s4d_params_kernel<<<pn / 256, 256, 0, stream>>>(PMID(0), PMID(1), PMID(2), PMID(3), PMID(4), P + 1 * pn);
  s4d_params_kernel<<<pn / 256, 256, 0, stream>>>(POUT(0), POUT(1), POUT(2), POUT(3), POUT(4), P + 2 * pn);
  s4d_params_kernel<<<pn / 256, 256, 0, stream>>>(POUT(0), POUT(1), POUT(2), POUT(10), POUT(11), P + 3 * pn);

  hipMemsetAsync(zA, 0, S * sizeof(float), stream);
  ln_rows_kernel<<<ROWS, CC, 0, stream>>>(x, PIN(8), PIN(9), xln, nullptr);  // LN(x), loop-invariant

  // ---- DEQ fixed-point loop ----
  for (int it = 0; it < NITER; ++it) {
    float* zin  = (it & 1) ? zB : zA;
    float* zout = (it & 1) ? zA : zB;

    // feed = LN(z @ Wf + bf) * sigmoid(gamma)
    gemm_wmma_f32_kernel<<<G1, 128, 0, stream>>>(zin, Wf, bf, tmp1, ROWS, CC, CC);
    ln_rows_kernel<<<ROWS, CC, 0, stream>>>(tmp1, lnf_g, lnf_b, feed, gamma);

    // ---- input block: u1 = LN(x) - feed ----
    s4d_conv_kernel<<<BB * CC, 32, 0, stream>>>(xln, feed, P + 0 * pn, nullptr, PIN(5), yact, nullptr);
    ln_rows_kernel<<<ROWS, CC, 0, stream>>>(yact, PIN(8), PIN(9), ynrm, nullptr);
    gemm_wmma_f32_kernel<<<G2, 128, 0, stream>>>(ynrm, PIN(6), PIN(7), zg, ROWS, 2 * CC, CC);
    glu_combine_kernel<<<(int)(S / 256), 256, 0, stream>>>(zg, yact, ybuf);

    // ---- middle block ----
    ln_rows_kernel<<<ROWS, CC, 0, stream>>>(ybuf, PMID(8), PMID(9), ubuf, nullptr);
    s4d_conv_kernel<<<BB * CC, 32, 0, stream>>>(ubuf, nullptr, P + 1 * pn, nullptr, PMID(5), yact, nullptr);
    ln_rows_kernel<<<ROWS, CC, 0, stream>>>(yact, PMID(8), PMID(9), ynrm, nullptr);
    gemm_wmma_f32_kernel<<<G2, 128, 0, stream>>>(ynrm, PMID(6), PMID(7), zg, ROWS, 2 * CC, CC);
    glu_combine_kernel<<<(int)(S / 256), 256, 0, stream>>>(zg, yact, ybuf);

    // ---- output block (emits next feedback state via G kernel) ----
    ln_rows_kernel<<<ROWS, CC, 0, stream>>>(ybuf, POUT(8), POUT(9), ubuf, nullptr);
    s4d_conv_kernel<<<BB * CC, 32, 0, stream>>>(ubuf, nullptr, P + 2 * pn, P + 3 * pn, POUT(5), yact, zout);
    ln_rows_kernel<<<ROWS, CC, 0, stream>>>(yact, POUT(8), POUT(9), ynrm, nullptr);
    gemm_wmma_f32_kernel<<<G2, 128, 0, stream>>>(ynrm, POUT(6), POUT(7), zg, ROWS, 2 * CC, CC);
    glu_combine_kernel<<<(int)(S / 256), 256, 0, stream>>>(zg, yact, ybuf);
  }

  // After 12 iterations: y = ybuf, z_star = zA (iter 11 writes zA)
  hipMemcpyAsync(d_out, ybuf, S * sizeof(float), hipMemcpyDeviceToDevice, stream);
  norm_ratio_kernel<<<BB * CC, 256, 0, stream>>>(x, zA, ratios);
  mean_ratio_kernel<<<1, 256, 0, stream>>>(ratios, (float*)d_out + S);

  #undef PIN
  #undef PMID
  #undef POUT
}